// grouped_fc_layer_47244640256575
// MI455X (gfx1250) — compile-verified
//
#include <hip/hip_runtime.h>

// out[b,g] = sum_l x[b,g,l] * W[l,g] + bias[g]
// B=8192, G=8, L=4096, fp32. Memory-bound: 1 GiB of x @ 23.3 TB/s ~ 46 us floor.
// Strategy: V_WMMA_F32_16X16X4_F32 per 16-row tile (fixed g), B-matrix = weight
// chunk broadcast across all columns, so every column of D holds the row dot
// product. x is loaded exactly once; weight column staged in LDS per block.

typedef __attribute__((ext_vector_type(2))) float v2f;
typedef __attribute__((ext_vector_type(8))) float v8f;

#define GROUPS 8
#define LEN 4096
#define ROWS_PER_WAVE 16
#define WAVES_PER_BLOCK 8
#define THREADS (WAVES_PER_BLOCK * 32)          // wave32
#define ROWS_PER_BLOCK (ROWS_PER_WAVE * WAVES_PER_BLOCK)  // 128 b-rows per block

__global__ __launch_bounds__(THREADS)
void grouped_fc_diag_wmma(const float* __restrict__ x,
                          const float* __restrict__ w,     // (L, G) after squeeze
                          const float* __restrict__ bias,  // (G,)
                          float* __restrict__ out) {       // (B, G)
  __shared__ float w_s[LEN];

  const int g = blockIdx.y;

  // Stage weight column g into LDS (de-stride the (L,G) layout). Weight is
  // 128 KB total -> L2 resident; this 16 KB gather amortizes over 2 MB of x.
  for (int i = threadIdx.x; i < LEN; i += THREADS)
    w_s[i] = w[(size_t)i * GROUPS + g];
  __syncthreads();

  const int lane = threadIdx.x & 31;
  const int wave = threadIdx.x >> 5;
  const int half = lane >> 4;   // K-half select: lanes 0-15 -> K{0,1}, 16-31 -> K{2,3}
  const int m    = lane & 15;   // row (M) within the 16-row tile
  const int tileB = blockIdx.x * ROWS_PER_BLOCK + wave * ROWS_PER_WAVE;

  // Per-lane base: row (tileB+m) of x for group g, offset to this lane's K-half.
  const float* xr = x + ((size_t)(tileB + m) * GROUPS + g) * (size_t)LEN + half * 2;
  const float* ws = w_s + half * 2;

  // A layout (16x4 f32): VGPR0 = K0 (lanes 0-15) / K2 (lanes 16-31),
  //                      VGPR1 = K1 / K3  -> one float2 per lane.
  // B layout mirrors the half-split; since B[k][j] = w[l0+k] for every column j,
  // each lane supplies {w[l0+2h], w[l0+2h+1]} -> one ds_load_b64.
  v8f c0 = {};
  v8f c1 = {};
#pragma unroll 4
  for (int l0 = 0; l0 < LEN; l0 += 8) {
    v2f a0 = *(const v2f*)(xr + l0);
    v2f b0 = *(const v2f*)(ws + l0);
    c0 = __builtin_amdgcn_wmma_f32_16x16x4_f32(
        /*neg_a=*/false, a0, /*neg_b=*/false, b0,
        /*c_mod=*/(short)0, c0, /*reuse_a=*/false, /*reuse_b=*/false);

    v2f a1 = *(const v2f*)(xr + l0 + 4);
    v2f b1 = *(const v2f*)(ws + l0 + 4);
    c1 = __builtin_amdgcn_wmma_f32_16x16x4_f32(
        false, a1, false, b1, (short)0, c1, false, false);
  }

  v8f d = c0 + c1;  // all N columns equal the row dot-product

  // C/D layout: VGPR r holds M=r (lanes 0-15) and M=r+8 (lanes 16-31).
  // Lane 0 writes rows 0..7, lane 16 writes rows 8..15.
  const float bg = bias[g];
  if (m == 0) {
    const int rbase = tileB + half * 8;
#pragma unroll
    for (int r = 0; r < 8; ++r)
      out[(size_t)(rbase + r) * GROUPS + g] = d[r] + bg;
  }
}

extern "C" void kernel_launch(void* const* d_in, const int* in_sizes, int n_in,
                              void* d_out, int out_size, void* d_ws, size_t ws_size,
                              hipStream_t stream) {
  (void)n_in; (void)out_size; (void)d_ws; (void)ws_size;
  const float* x    = (const float*)d_in[0];   // (B, G, L) f32
  const float* w    = (const float*)d_in[1];   // (1, L, G) f32
  const float* bias = (const float*)d_in[2];   // (G,) f32
  float*       out  = (float*)d_out;           // (B, G) f32

  const int Btot = in_sizes[0] / (GROUPS * LEN);  // 8192
  dim3 grid(Btot / ROWS_PER_BLOCK, GROUPS, 1);    // (64, 8)
  dim3 block(THREADS, 1, 1);
  hipLaunchKernelGGL(grouped_fc_diag_wmma, grid, block, 0, stream, x, w, bias, out);
}